// ResidualVoxelToPointDecoder_81381040325407
// MI455X (gfx1250) — compile-verified
//
#include <hip/hip_runtime.h>

typedef __attribute__((ext_vector_type(2))) float v2f;
typedef __attribute__((ext_vector_type(8))) float v8f;

#define NPTS   4096
#define NBATCH 2
#define MROWS  (NBATCH * NPTS)
#define BN_EPS 1e-3f

// ---------------------------------------------------------------------------
// coords output: [B*Np, 4] = (batch_idx, x, y, z)
// ---------------------------------------------------------------------------
__global__ void coords_kernel(const float* __restrict__ kp, float* __restrict__ out) {
    int i = blockIdx.x * blockDim.x + threadIdx.x;
    if (i >= MROWS) return;
    out[i * 4 + 0] = (float)(i / NPTS);
    out[i * 4 + 1] = kp[i * 3 + 0];
    out[i * 4 + 2] = kp[i * 3 + 1];
    out[i * 4 + 3] = kp[i * 3 + 2];
}

// ---------------------------------------------------------------------------
// Top-3 NN + inverse-distance interpolation. One thread per keypoint,
// voxel centers tiled through LDS (reused by 128 points per block).
// coords are [z,y,x] ints; centers = (c+0.5)*voxel*ds + pcr_min (xyz).
// ---------------------------------------------------------------------------
template <int C>
__global__ void interp_kernel(const float* __restrict__ kp,
                              const int*   __restrict__ coords,
                              const float* __restrict__ feats,
                              float*       __restrict__ out,
                              int Nv, float ds) {
    const int TILE = 256;
    __shared__ float sx[TILE], sy[TILE], sz[TILE], s2[TILE];

    int b = blockIdx.y;
    int p = blockIdx.x * blockDim.x + threadIdx.x;   // 0..NPTS-1 (exact)
    int g = b * NPTS + p;

    float px = kp[g * 3 + 0];
    float py = kp[g * 3 + 1];
    float pz = kp[g * 3 + 2];
    float p2 = px * px + py * py + pz * pz;

    float d0 = 3.4e38f, d1 = 3.4e38f, d2 = 3.4e38f;
    int   i0 = 0, i1 = 0, i2 = 0;

    const int* cb = coords + (size_t)b * Nv * 3;

    for (int t = 0; t < Nv; t += TILE) {
        int n = min(TILE, Nv - t);
        for (int j = threadIdx.x; j < n; j += blockDim.x) {
            int zc = cb[(t + j) * 3 + 0];
            int yc = cb[(t + j) * 3 + 1];
            int xc = cb[(t + j) * 3 + 2];
            float cx = ((float)xc + 0.5f) * (0.05f * ds) + 0.0f;
            float cy = ((float)yc + 0.5f) * (0.05f * ds) - 40.0f;
            float cz = ((float)zc + 0.5f) * (0.10f * ds) - 3.0f;
            sx[j] = cx; sy[j] = cy; sz[j] = cz;
            s2[j] = cx * cx + cy * cy + cz * cz;
        }
        __syncthreads();
        for (int j = 0; j < n; ++j) {
            float dot = px * sx[j] + py * sy[j] + pz * sz[j];
            float d   = p2 + s2[j] - 2.0f * dot;     // same formula as reference
            int   idx = t + j;
            if (d < d2) {
                if (d < d1) {
                    d2 = d1; i2 = i1;
                    if (d < d0) { d1 = d0; i1 = i0; d0 = d; i0 = idx; }
                    else        { d1 = d;  i1 = idx; }
                } else { d2 = d; i2 = idx; }
            }
        }
        __syncthreads();
    }

    float w0 = 1.0f / (fmaxf(d0, 0.0f) + 1e-8f);
    float w1 = 1.0f / (fmaxf(d1, 0.0f) + 1e-8f);
    float w2 = 1.0f / (fmaxf(d2, 0.0f) + 1e-8f);
    float inv = 1.0f / (w0 + w1 + w2);
    w0 *= inv; w1 *= inv; w2 *= inv;

    const float* fb = feats + (size_t)b * Nv * C;
    const float* f0 = fb + (size_t)i0 * C;
    const float* f1 = fb + (size_t)i1 * C;
    const float* f2 = fb + (size_t)i2 * C;
    float* ob = out + (size_t)g * C;
#pragma unroll 4
    for (int c = 0; c < C; ++c)
        ob[c] = w0 * f0[c] + w1 * f1[c] + w2 * f2[c];
}

// ---------------------------------------------------------------------------
// WMMA fp32 GEMM with fused BN (+optional residual add) (+optional ReLU).
//   OUT[M,N] = act( (X[M,K] @ W[K,N]) * s + t  [+ ADD] ),  s/t from BN stats.
// One wave -> 16 output rows x all N columns (NT = N/16 accumulators).
// Fragment layouts per CDNA5 ISA 7.12.2:
//   A 16x4:  lanes 0-15 hold K={0,1}, lanes 16-31 hold K={2,3}; M = lane&15
//   B 4x16:  lanes 0-15 hold K={0,1}, lanes 16-31 hold K={2,3}; N = lane&15
//   C/D:     VGPR i = row (8*half + i), col = lane&15
// ---------------------------------------------------------------------------
template <int NT>
__global__ void gemm_bn_act(const float* __restrict__ X, const float* __restrict__ W,
                            const float* __restrict__ gma, const float* __restrict__ bta,
                            const float* __restrict__ mea, const float* __restrict__ vra,
                            const float* __restrict__ ADD, float* __restrict__ OUT,
                            int K, int relu) {
    const int N = NT * 16;
    int wid  = (blockIdx.x * blockDim.x + threadIdx.x) >> 5;
    int lane = threadIdx.x & 31;
    int half = lane >> 4;
    int r    = lane & 15;
    int row0 = wid * 16;
    if (row0 >= MROWS) return;   // wave-uniform

    v8f acc[NT] = {};

    const float* xp = X + (size_t)(row0 + r) * K + 2 * half;
    const float* wp = W + (size_t)(2 * half) * N + r;

    for (int k0 = 0; k0 < K; k0 += 4) {
        v2f a;
        a[0] = xp[k0];
        a[1] = xp[k0 + 1];
#pragma unroll
        for (int t = 0; t < NT; ++t) {
            const float* wq = wp + (size_t)k0 * N + t * 16;
            v2f bf;
            bf[0] = wq[0];
            bf[1] = wq[N];
            acc[t] = __builtin_amdgcn_wmma_f32_16x16x4_f32(
                false, a, false, bf, (short)0, acc[t], false, false);
        }
    }

#pragma unroll
    for (int t = 0; t < NT; ++t) {
        int col  = t * 16 + r;
        float s  = gma[col] * rsqrtf(vra[col] + BN_EPS);
        float bb = bta[col] - mea[col] * s;
#pragma unroll
        for (int i = 0; i < 8; ++i) {
            int row = row0 + half * 8 + i;
            float y = acc[t][i] * s + bb;
            if (ADD)  y += ADD[(size_t)row * N + col];
            if (relu) y = fmaxf(y, 0.0f);
            OUT[(size_t)row * N + col] = y;
        }
    }
}

// ---------------------------------------------------------------------------
extern "C" void kernel_launch(void* const* d_in, const int* in_sizes, int n_in,
                              void* d_out, int out_size, void* d_ws, size_t ws_size,
                              hipStream_t stream) {
    (void)in_sizes; (void)n_in; (void)out_size; (void)ws_size;

    const float* kp  = (const float*)d_in[0];
    const float* vf1 = (const float*)d_in[1];
    const int*   vc1 = (const int*)  d_in[2];
    const float* vf2 = (const float*)d_in[3];
    const int*   vc2 = (const int*)  d_in[4];
    const float* vf3 = (const float*)d_in[5];
    const int*   vc3 = (const int*)  d_in[6];
    const float* vf4 = (const float*)d_in[7];
    const int*   vc4 = (const int*)  d_in[8];

    // params flattened depth-first in dict insertion order:
    // b4, b3, b2, b1 (each: W1, bn1{g,b,m,v}, W2, bn2{..}, Wd, bnd{..} = 15 arrays), out{W, bn{g,b,m,v}}
    const int PB = 9;
    auto prm = [&](int i) { return (const float*)d_in[PB + i]; };

    float* ws    = (float*)d_ws;
    const size_t SLOT = (size_t)MROWS * 64;   // 524288 floats
    float* bufI  = ws;
    float* bufH1 = ws + SLOT;
    float* bufH2 = ws + 2 * SLOT;
    float* bufFA = ws + 3 * SLOT;
    float* bufFB = ws + 4 * SLOT;

    float* outC = (float*)d_out;               // coords [8192,4]
    float* outF = (float*)d_out + MROWS * 4;   // features [8192,128]

    coords_kernel<<<(MROWS + 255) / 256, 256, 0, stream>>>(kp, outC);

    const int GEMM_GRID = MROWS / 16 / 8;      // 512 waves, 8 waves/block

    auto gemm64 = [&](const float* X, int K, const float* Wm,
                      const float* g, const float* be, const float* mn, const float* vr,
                      const float* ADD, float* OUT, int relu) {
        gemm_bn_act<4><<<GEMM_GRID, 256, 0, stream>>>(X, Wm, g, be, mn, vr, ADD, OUT, K, relu);
    };

    // res_block: h1 = relu(bn1(res@W1)); h2 = bn2(h1@W2); out = relu(bnd(bot@Wd) + h2)
    auto res_block = [&](const float* res, int Kres, const float* bot, int blk, float* outp) {
        const float* W1 = prm(blk * 15 + 0);
        const float* g1 = prm(blk * 15 + 1), *b1 = prm(blk * 15 + 2);
        const float* m1 = prm(blk * 15 + 3), *v1 = prm(blk * 15 + 4);
        const float* W2 = prm(blk * 15 + 5);
        const float* g2 = prm(blk * 15 + 6), *b2 = prm(blk * 15 + 7);
        const float* m2 = prm(blk * 15 + 8), *v2 = prm(blk * 15 + 9);
        const float* Wd = prm(blk * 15 + 10);
        const float* gd = prm(blk * 15 + 11), *bd = prm(blk * 15 + 12);
        const float* md = prm(blk * 15 + 13), *vd = prm(blk * 15 + 14);
        gemm64(res,   Kres, W1, g1, b1, m1, v1, nullptr, bufH1, 1);
        gemm64(bufH1, 64,   W2, g2, b2, m2, v2, nullptr, bufH2, 0);
        gemm64(bot,   64,   Wd, gd, bd, md, vd, bufH2,   outp,  1);
    };

    dim3 igrid(NPTS / 128, NBATCH);

    // stage c4 (block b4 = params index 0): res == bot == interp(c4)
    interp_kernel<64><<<igrid, 128, 0, stream>>>(kp, vc4, vf4, bufI, 2048, 8.0f);
    res_block(bufI, 64, bufI, 0, bufFA);

    // stage c3 (b3 = 1)
    interp_kernel<64><<<igrid, 128, 0, stream>>>(kp, vc3, vf3, bufI, 4096, 4.0f);
    res_block(bufI, 64, bufFA, 1, bufFB);

    // stage c2 (b2 = 2), C_lat = 32
    interp_kernel<32><<<igrid, 128, 0, stream>>>(kp, vc2, vf2, bufI, 8192, 2.0f);
    res_block(bufI, 32, bufFB, 2, bufFA);

    // stage c1 (b1 = 3), C_lat = 16
    interp_kernel<16><<<igrid, 128, 0, stream>>>(kp, vc1, vf1, bufI, 16384, 1.0f);
    res_block(bufI, 16, bufFA, 3, bufFB);

    // final: relu(bn(f1 @ W[64,128])) -> d_out
    const float* Wo = prm(60);
    const float* go = prm(61), *bo = prm(62), *mo = prm(63), *vo = prm(64);
    gemm_bn_act<8><<<GEMM_GRID, 256, 0, stream>>>(bufFB, Wo, go, bo, mo, vo,
                                                  nullptr, outF, 64, 1);
}